// MomentumLora_62809601736885
// MI455X (gfx1250) — compile-verified
//
#include <hip/hip_runtime.h>

typedef __attribute__((ext_vector_type(2))) float v2f;
typedef __attribute__((ext_vector_type(8))) float v8f;

#define DIM   768
#define RANK  16
#define NTOT  2304
#define SCALE (1.0f / 16.0f)

// One block (4 wave32) processes one 16-row tile of x.
// Stage 0: async-copy X tile (48 KB, contiguous) global -> LDS (ASYNCcnt path)
// Stage 1: T[16x16] per q/k/v = X_tile @ A  (K=768 split across 4 waves, LDS reduce)
// Stage 2: out_tile[16x2304] = (T * SCALE) @ B  (144 N-tiles split across 4 waves),
//          streamed out with non-temporal stores (write-once, > L2 capacity).
__global__ __launch_bounds__(128)
void lora_fused_wmma_kernel(const float* __restrict__ x,
                            const float* __restrict__ qA, const float* __restrict__ qB,
                            const float* __restrict__ kA, const float* __restrict__ kB,
                            const float* __restrict__ vA, const float* __restrict__ vB,
                            const int*   __restrict__ depth_id,
                            float*       __restrict__ out)
{
    __shared__ float ldsX[16 * DIM];     // 48 KB: X tile, row-major
    __shared__ float ldsT[4 * 3 * 256];  // 12 KB: per-wave partial T's; final T in [0,768)

    const int tid   = threadIdx.x;
    const int lane  = tid & 31;
    const int wv    = tid >> 5;            // wave id 0..3
    const int m     = lane & 15;           // row (A-frag) / col (B-frag) index
    const int khalf = (lane >> 4) << 1;    // 0 or 2: K sub-offset per half-wave
    const int Mbase = (lane >> 4) << 3;    // 0 or 8: C/D row base per half-wave

    const long r0 = (long)blockIdx.x * 16; // first of 16 rows of x handled here

    const int d = depth_id[0];
    const size_t offA = (size_t)d * DIM * RANK;
    const size_t offB = (size_t)d * RANK * DIM;
    const float* Aq = qA + offA; const float* Ak = kA + offA; const float* Av = vA + offA;
    const float* Bq = qB + offB; const float* Bk = kB + offB; const float* Bv = vB + offB;

    // ---- stage 0: async global->LDS copy of the X tile ----
    // 16 rows of a row-major [25216,768] matrix form one contiguous 48 KB span.
    // 128 threads x 24 iterations x 16 B = 49152 B. GVS addressing:
    //   lds[VDST_vgpr] = mem[SADDR_sgpr64 + VADDR_vgpr32]; tracked by ASYNCcnt.
    // Low 32 bits of a flat shared-aperture address == LDS byte offset.
    {
        const float* src = x + (size_t)r0 * DIM;           // uniform 64-bit base (SGPRs)
        unsigned ldsA = (unsigned)(size_t)&ldsX[0] + (unsigned)tid * 16u;
        unsigned gOff = (unsigned)tid * 16u;
#pragma unroll
        for (int i = 0; i < 24; ++i) {
            asm volatile("global_load_async_to_lds_b128 %0, %1, %2"
                         :: "v"(ldsA), "v"(gOff), "s"(src)
                         : "memory");
            ldsA += 2048u;   // 128 threads * 16 B per sweep
            gOff += 2048u;
        }
        asm volatile("s_wait_asynccnt 0x0" ::: "memory");
    }
    __syncthreads();

    // ---- stage 1: per-wave partial T = X(:, k-slice) @ A(k-slice, :) ----
    v8f accQ = {}, accK = {}, accV = {};
    const int kbeg = wv * 192;
#pragma unroll 4
    for (int k = kbeg; k < kbeg + 192; k += 4) {
        const int kk = k + khalf;
        // A-operand frag: X[m][kk], X[m][kk+1]  (float2 from LDS)
        v2f a = *(const v2f*)&ldsX[m * DIM + kk];
        // B-operand frags: A_lora[kk][m], A_lora[kk+1][m]  (row stride = 16 floats, L2-hot)
        v2f bq = { Aq[kk * RANK + m], Aq[(kk + 1) * RANK + m] };
        v2f bk = { Ak[kk * RANK + m], Ak[(kk + 1) * RANK + m] };
        v2f bv = { Av[kk * RANK + m], Av[(kk + 1) * RANK + m] };
        accQ = __builtin_amdgcn_wmma_f32_16x16x4_f32(false, a, false, bq, (short)0, accQ, false, false);
        accK = __builtin_amdgcn_wmma_f32_16x16x4_f32(false, a, false, bk, (short)0, accK, false, false);
        accV = __builtin_amdgcn_wmma_f32_16x16x4_f32(false, a, false, bv, (short)0, accV, false, false);
    }

    // spill per-wave partials (C/D layout: VGPR v, lane L -> T[Mbase+v][m])
    {
        float* p = &ldsT[wv * 768];
#pragma unroll
        for (int v = 0; v < 8; ++v) {
            const int idx = (Mbase + v) * 16 + m;
            p[idx]       = accQ[v];
            p[256 + idx] = accK[v];
            p[512 + idx] = accV[v];
        }
    }
    __syncthreads();
    // reduce 4 wave partials: 768 elems / 128 threads = 6 each (disjoint read/write sets)
#pragma unroll
    for (int j = 0; j < 6; ++j) {
        const int e = tid * 6 + j;
        ldsT[e] = ldsT[e] + ldsT[768 + e] + ldsT[1536 + e] + ldsT[2304 + e];
    }
    __syncthreads();

    // ---- stage 2: out = (T * SCALE) @ B ; 144 N-tiles of 16, 36 per wave ----
    for (int t = wv; t < 144; t += 4) {
        const int mat  = t / 48;          // 0=q, 1=k, 2=v (uniform within wave)
        const int ncol = (t % 48) * 16;   // column offset within that matrix
        const float* Bm = (mat == 0) ? Bq : (mat == 1) ? Bk : Bv;
        const float* Tm = &ldsT[mat * 256];
        v8f acc = {};
#pragma unroll
        for (int c = 0; c < 4; ++c) {
            const int kk = 4 * c + khalf;
            // A-frag from T (fold LoRA scaling here)
            v2f a = { Tm[m * 16 + kk] * SCALE, Tm[m * 16 + kk + 1] * SCALE };
            // B-frag from B_lora [16 x 768], row stride 768 floats (L2-hot)
            v2f b = { Bm[kk * DIM + ncol + m], Bm[(kk + 1) * DIM + ncol + m] };
            acc = __builtin_amdgcn_wmma_f32_16x16x4_f32(false, a, false, b, (short)0, acc, false, false);
        }
        // store 16x16 tile: VGPR v, lane L -> out[r0 + Mbase + v][mat*768 + ncol + m]
        // Non-temporal: output is write-once and larger than L2 (232 MB > 192 MB);
        // keep it from evicting the hot A/B factors and streaming x lines.
        float* o = out + (size_t)(r0 + Mbase) * NTOT + (size_t)mat * DIM + ncol + m;
#pragma unroll
        for (int v = 0; v < 8; ++v)
            __builtin_nontemporal_store(acc[v], o + (size_t)v * NTOT);
    }
}

extern "C" void kernel_launch(void* const* d_in, const int* in_sizes, int n_in,
                              void* d_out, int out_size, void* d_ws, size_t ws_size,
                              hipStream_t stream) {
    (void)in_sizes; (void)n_in; (void)out_size; (void)d_ws; (void)ws_size;
    const float* x  = (const float*)d_in[0];
    const float* qA = (const float*)d_in[1];
    const float* qB = (const float*)d_in[2];
    const float* kA = (const float*)d_in[3];
    const float* kB = (const float*)d_in[4];
    const float* vA = (const float*)d_in[5];
    const float* vB = (const float*)d_in[6];
    const int*   dd = (const int*)d_in[7];
    float* out = (float*)d_out;

    const int rowTiles = (128 * 197) / 16;  // 25216 rows -> 1576 tiles, exact
    hipLaunchKernelGGL(lora_fused_wmma_kernel, dim3(rowTiles), dim3(128), 0, stream,
                       x, qA, qB, kA, kB, vA, vB, dd, out);
}